// NoncommutativeKolmogorovArnoldRigorousOperator_32418413150389
// MI455X (gfx1250) — compile-verified
//
#include <hip/hip_runtime.h>
#include <hip/hip_bf16.h>
#include <math.h>

#define DIM 2048
#define THETA 1e-18
#define REG_EPS 1e-15
#define ZETA2 1.6449340668482264
#define PI_D 3.14159265358979323846
#define EXP_M02 0.81873075307798185867    // exp(-0.2)
#define SCALE_BACK 1e-36                  // undo E' = E * 1e18 twice

typedef __attribute__((ext_vector_type(2))) float v2f;
typedef __attribute__((ext_vector_type(8))) float v8f;

// first 50 primes (all <= 229), matching reference PRIMES[:50]
__constant__ int PRIMES50[50] = {
    2,3,5,7,11,13,17,19,23,29,31,37,41,43,47,53,59,61,67,71,
    73,79,83,89,97,101,103,107,109,113,127,131,137,139,149,151,157,163,167,173,
    179,181,191,193,197,199,211,223,227,229};

__device__ __forceinline__ bool is_prime50(int v) {
    if (v < 2 || v > 229) return false;
    bool hit = false;
#pragma unroll
    for (int t = 0; t < 50; ++t) hit |= (PRIMES50[t] == v);
    return hit;
}

// ---------------- fp64 closed-form band entries ----------------

// E(r,c): off-band part of Arnold map A = I + E (real symmetric).
__device__ __forceinline__ double e64(int r, int c) {
    if (r < 0 || c < 0 || r >= DIM || c >= DIM) return 0.0;
    int d = r - c; if (d < 0) d = -d;
    if (d == 1) {
        int m = (r < c) ? r : c;
        return THETA * sin(PI_D * (2.0 * m + 1.0) / (double)DIM);
    }
    if (d == 2) return THETA * THETA * EXP_M02;
    return 0.0;
}

// M_h = hermitized zeta/prime core: real diag Re(n^-s) + zeta2 prime corr,
// pure-imaginary prime entries at (p-1,p)/(p,p-1).
__device__ __forceinline__ void mh64(int r, int c, double si, double* re, double* im) {
    *re = 0.0; *im = 0.0;
    if (r < 0 || c < 0 || r >= DIM || c >= DIM) return;
    if (r == c) {
        double L = log((double)(r + 1));
        double v = exp(-0.5 * L) * cos(si * L);
        if (is_prime50(r + 1)) v += (THETA * L) * ZETA2;
        *re = v;
    } else {
        int d = r - c; if (d < 0) d = -d;
        if (d == 1) {
            int hi = (r > c) ? r : c;           // larger index == prime value p
            if (is_prime50(hi)) {
                double corr = 0.5 * THETA * log((double)hi);
                *im = (c > r) ? corr : -corr;   // (p-1,p)=+i/2, (p,p-1)=-i/2
            }
        }
    }
}

// G: noncommutative algebra (already Hermitian)
__device__ __forceinline__ void g64(int r, int c, double* re, double* im) {
    *re = 0.0; *im = 0.0;
    if (r < 0 || c < 0 || r >= DIM || c >= DIM) return;
    int d = c - r;
    if (d == 1)       *im =  THETA;
    else if (d == -1) *im = -THETA;
    else if (d == 2 || d == -2) {
        int m = (r < c) ? r : c;
        *re = THETA * THETA * exp(-(double)m / 100.0);
    }
}

// ---------------- fp32 (scaled) versions for the WMMA path ----------------

__device__ __forceinline__ float e32s(int r, int c) {   // E * 1e18
    if (r < 0 || c < 0 || r >= DIM || c >= DIM) return 0.0f;
    int d = r - c; if (d < 0) d = -d;
    if (d == 1) {
        int m = (r < c) ? r : c;
        return __sinf((float)PI_D * (2.0f * (float)m + 1.0f) / (float)DIM);
    }
    if (d == 2) return (float)(THETA * EXP_M02);        // theta^2*e^-.2*1e18
    return 0.0f;
}

__device__ __forceinline__ void mh32(int r, int c, float si, float* re, float* im) {
    *re = 0.0f; *im = 0.0f;
    if (r < 0 || c < 0 || r >= DIM || c >= DIM) return;
    if (r == c) {
        float L = __logf((float)(r + 1));
        float v = __expf(-0.5f * L) * __cosf(si * L);
        if (is_prime50(r + 1)) v += (float)(THETA * ZETA2) * L;
        *re = v;
    } else {
        int d = r - c; if (d < 0) d = -d;
        if (d == 1) {
            int hi = (r > c) ? r : c;
            if (is_prime50(hi)) {
                float corr = 0.5f * (float)THETA * __logf((float)hi);
                *im = (c > r) ? corr : -corr;
            }
        }
    }
}

// ---------------- kernel 1: zero fill via async LDS->global B128 stores ----------------
// 64 MiB of structural zeros is the roofline-dominant cost (~2.9 us @ 23.3 TB/s).
// Stage a 4 KB zero tile in LDS, then drain it with GLOBAL_STORE_ASYNC_FROM_LDS_B128
// (ASYNCcnt-tracked); the wave only generates addresses.

#define ZF_BLOCKS 1024
#define ZF_CHUNKS ((size_t)DIM * DIM * 16 / 4096)   // 16384 x 4KB chunks

__global__ void __launch_bounds__(256) zero_fill_async_kernel(double* __restrict__ out) {
    __shared__ float4 zbuf[256];                    // 4 KB of zeros
    int tid = threadIdx.x;
    float4 z; z.x = 0.f; z.y = 0.f; z.z = 0.f; z.w = 0.f;
    zbuf[tid] = z;
    __syncthreads();                                // LDS visible before async engine reads

    // LDS byte offset of this lane's 16-byte zero source (low 32 bits of the
    // generic shared address == LDS offset, ISA 10.2 aperture rule)
    unsigned lds_addr = (unsigned)(unsigned long long)(void*)&zbuf[tid];
    unsigned long long base = (unsigned long long)out + (unsigned long long)tid * 16ull;

    for (size_t c = blockIdx.x; c < ZF_CHUNKS; c += ZF_BLOCKS) {
        unsigned long long gaddr = base + c * 4096ull;
        asm volatile("global_store_async_from_lds_b128 %0, %1, off"
                     :: "v"(gaddr), "v"(lds_addr) : "memory");
    }
    asm volatile("s_wait_asynccnt 0x0" ::: "memory");
}

// ---------------- kernel 2: fp64 band (M_h + E*M_h + M_h*E^T + |s|G + reg) ----------------

__global__ void band_kernel(const double* __restrict__ s_real,
                            const double* __restrict__ s_imag,
                            double* __restrict__ out) {
    int t = blockIdx.x * blockDim.x + threadIdx.x;
    if (t >= DIM * 11) return;
    int i = t / 11;
    int j = i + (t % 11) - 5;
    if (j < 0 || j >= DIM) return;

    double sr = s_real[0], si = s_imag[0];
    double sabs = sqrt(sr * sr + si * si);

    double re, im, tre, tim;
    mh64(i, j, si, &re, &im);
#pragma unroll
    for (int d = -2; d <= 2; ++d) {
        if (d == 0) continue;
        double ea = e64(i, i + d);                  // (E*M_h)(i,j)
        if (ea != 0.0) { mh64(i + d, j, si, &tre, &tim); re += ea * tre; im += ea * tim; }
        double eb = e64(j, j + d);                  // (M_h*E^T)(i,j)
        if (eb != 0.0) { mh64(i, j + d, si, &tre, &tim); re += eb * tre; im += eb * tim; }
    }
    g64(i, j, &tre, &tim);
    re += sabs * tre; im += sabs * tim;
    if (i == j) re += REG_EPS;

    size_t off = ((size_t)i * DIM + j) * 2;
    out[off + 0] = re;
    out[off + 1] = im;
}

// ---------------- kernel 3: second-order E*M_h*E^T via v_wmma_f32_16x16x4_f32 ----------------
// One wave per 16-row block. Window of 32 band columns covers all nonzero k,l.
// |term| <= 1e-36 absolute, so fp32 relative error contributes <= ~1e-43.

__global__ void __launch_bounds__(32) wmma_second_order_kernel(
        const double* __restrict__ s_imag, double* __restrict__ out) {
    __shared__ float Tr[16 * 32];
    __shared__ float Ti[16 * 32];

    int bi = blockIdx.x;            // 0..127
    int w0 = 16 * bi - 8;           // band window base
    int lane = threadIdx.x;
    int mrow = lane & 15;
    int half = lane >> 4;           // 0: lanes 0-15, 1: lanes 16-31
    float si = (float)s_imag[0];

    // T = E'(tile-rows, window) * M_h(window, window) : 16x32, real+imag planes
    for (int tcol = 0; tcol < 2; ++tcol) {
        v8f accr = {0.f,0.f,0.f,0.f,0.f,0.f,0.f,0.f};
        v8f acci = {0.f,0.f,0.f,0.f,0.f,0.f,0.f,0.f};
        int n = 16 * tcol + (lane & 15);            // local column in window
        for (int kk = 0; kk < 8; ++kk) {
            int k0 = 4 * kk + 2 * half;             // ISA A-layout: VGPRv holds K = 2*half+v
            v2f a;
            a[0] = e32s(16 * bi + mrow, w0 + k0);
            a[1] = e32s(16 * bi + mrow, w0 + k0 + 1);
            float re0, im0, re1, im1;
            mh32(w0 + k0,     w0 + n, si, &re0, &im0);
            mh32(w0 + k0 + 1, w0 + n, si, &re1, &im1);
            v2f br; br[0] = re0; br[1] = re1;
            v2f bim; bim[0] = im0; bim[1] = im1;
            accr = __builtin_amdgcn_wmma_f32_16x16x4_f32(
                false, a, false, br, (short)0, accr, false, false);
            acci = __builtin_amdgcn_wmma_f32_16x16x4_f32(
                false, a, false, bim, (short)0, acci, false, false);
        }
        // D layout -> row-major LDS [m][l] for re-use as A operand
#pragma unroll
        for (int v = 0; v < 8; ++v) {
            int m = v + 8 * half;
            Tr[m * 32 + n] = accr[v];
            Ti[m * 32 + n] = acci[v];
        }
    }
    __syncthreads();

    // U(bj) = T(16x32) * E'(window, tile-bj cols)(32x16), for bj = bi-1..bi+1
    for (int dj = -1; dj <= 1; ++dj) {
        int bj = bi + dj;
        v8f ur = {0.f,0.f,0.f,0.f,0.f,0.f,0.f,0.f};
        v8f ui = {0.f,0.f,0.f,0.f,0.f,0.f,0.f,0.f};
        int n = lane & 15;
        for (int kk = 0; kk < 8; ++kk) {
            int l0 = 4 * kk + 2 * half;
            v2f ar;  ar[0]  = Tr[mrow * 32 + l0]; ar[1]  = Tr[mrow * 32 + l0 + 1];
            v2f aim; aim[0] = Ti[mrow * 32 + l0]; aim[1] = Ti[mrow * 32 + l0 + 1];
            v2f b;
            b[0] = e32s(w0 + l0,     16 * bj + n);   // E symmetric
            b[1] = e32s(w0 + l0 + 1, 16 * bj + n);
            ur = __builtin_amdgcn_wmma_f32_16x16x4_f32(
                false, ar, false, b, (short)0, ur, false, false);
            ui = __builtin_amdgcn_wmma_f32_16x16x4_f32(
                false, aim, false, b, (short)0, ui, false, false);
        }
        if (bj < 0 || bj >= DIM / 16) continue;      // wave-uniform
#pragma unroll
        for (int v = 0; v < 8; ++v) {
            int i = 16 * bi + v + 8 * half;
            int j = 16 * bj + n;
            size_t off = ((size_t)i * DIM + j) * 2;
            out[off + 0] += (double)ur[v] * SCALE_BACK;
            out[off + 1] += (double)ui[v] * SCALE_BACK;
        }
    }
}

extern "C" void kernel_launch(void* const* d_in, const int* in_sizes, int n_in,
                              void* d_out, int out_size, void* d_ws, size_t ws_size,
                              hipStream_t stream) {
    (void)in_sizes; (void)n_in; (void)out_size; (void)d_ws; (void)ws_size;
    const double* s_real = (const double*)d_in[0];
    const double* s_imag = (const double*)d_in[1];
    double* out = (double*)d_out;   // complex128 as interleaved (re, im) doubles

    // 1) 64 MiB structural-zero fill via async LDS->global stores (ASYNCcnt path)
    zero_fill_async_kernel<<<ZF_BLOCKS, 256, 0, stream>>>(out);
    // 2) exact fp64 band |i-j|<=5 (theta=1e-18 structure requires fp64)
    band_kernel<<<(DIM * 11 + 255) / 256, 256, 0, stream>>>(s_real, s_imag, out);
    // 3) second-order E*M_h*E^T via fp32 WMMA (scaled 1e18, error <= 1e-43 absolute)
    wmma_second_order_kernel<<<DIM / 16, 32, 0, stream>>>(s_imag, out);
}